// GRAFFNN_44839458570572
// MI455X (gfx1250) — compile-verified
//
#include <hip/hip_runtime.h>

#define N_NODES 100000
#define N_EDGES 1600000
#define IN_CH 16
#define HID 128
#define NUM_LAYERS 8

typedef __attribute__((ext_vector_type(2))) float v2f;
typedef __attribute__((ext_vector_type(8))) float v8f;

// ---------------- weighted degree + in-edge counts ----------------

__global__ void graff_deg_count(const long long* __restrict__ col,
                                const float* __restrict__ w,
                                float* __restrict__ deg,
                                int* __restrict__ cnt, int E) {
    int e = blockIdx.x * blockDim.x + threadIdx.x;
    if (e < E) {
        int c = (int)col[e];
        atomicAdd(&deg[c], w[e]);
        atomicAdd(&cnt[c], 1);
    }
}

__global__ void graff_dinv(const float* __restrict__ deg,
                           float* __restrict__ dinv, int n) {
    int i = blockIdx.x * blockDim.x + threadIdx.x;
    if (i < n) {
        float d = deg[i];
        dinv[i] = (d > 0.f) ? (1.0f / sqrtf(fmaxf(d, 1e-12f))) : 0.f;
    }
}

// ---------------- 3-pass exclusive scan of cnt -> starts ----------------

__global__ void graff_scan_block(const int* __restrict__ cnt,
                                 int* __restrict__ starts,
                                 int* __restrict__ blockSums, int n) {
    __shared__ int sh[256];
    const int t = threadIdx.x;
    const int i = blockIdx.x * 256 + t;
    int v = (i < n) ? cnt[i] : 0;
    sh[t] = v;
    __syncthreads();
    for (int off = 1; off < 256; off <<= 1) {
        int add = (t >= off) ? sh[t - off] : 0;
        __syncthreads();
        sh[t] += add;
        __syncthreads();
    }
    if (i < n) starts[i] = sh[t] - v;                 // exclusive within block
    if (t == 255) blockSums[blockIdx.x] = sh[255];
}

__global__ void graff_scan_tops(int* __restrict__ blockSums, int nb) {
    if (blockIdx.x == 0 && threadIdx.x == 0) {
        int run = 0;
        for (int b = 0; b < nb; ++b) { int t = blockSums[b]; blockSums[b] = run; run += t; }
    }
}

__global__ void graff_scan_add(int* __restrict__ starts,
                               const int* __restrict__ blockSums, int n) {
    int i = blockIdx.x * 256 + threadIdx.x;
    if (i < n) starts[i] += blockSums[blockIdx.x];
}

// ---------------- CSR fill (norm folded in): srcIdx / wcsr grouped by dst ----------------

__global__ void graff_fill(const long long* __restrict__ row,
                           const long long* __restrict__ col,
                           const float* __restrict__ ew,
                           const float* __restrict__ dinv,
                           const int* __restrict__ starts,
                           int* __restrict__ cursor,
                           int* __restrict__ srcIdx,
                           float* __restrict__ wcsr, int E) {
    int e = blockIdx.x * blockDim.x + threadIdx.x;
    if (e >= E) return;
    int r = (int)row[e], c = (int)col[e];
    int pos = starts[c] + atomicAdd(&cursor[c], 1);
    srcIdx[pos] = r;
    wcsr[pos] = dinv[r] * ew[e] * dinv[c];
}

__global__ void graff_wsym(const float* __restrict__ W,
                           float* __restrict__ Wsym) {
    int i = blockIdx.x * blockDim.x + threadIdx.x;   // 0..16383
    int r = i >> 7, c = i & 127;
    Wsym[i] = 0.5f * (W[i] + W[c * HID + r]);
}

// ---------------- fp32 WMMA GEMM: wave computes 16x64 strip via 4x V_WMMA_F32_16X16X4_F32 ----------------

__global__ void graff_wmma_gemm(const float* __restrict__ A,  // [M x K]
                                const float* __restrict__ B,  // [K x 128]
                                const float* __restrict__ bias,
                                float* __restrict__ C,        // [M x 128]
                                float* __restrict__ C2,       // nullable second dest
                                int M, int K) {
    const int lane  = threadIdx.x & 31;
    const int wave  = threadIdx.x >> 5;
    const int mBase = (blockIdx.x * 8 + wave) * 16;
    if (mBase >= M) return;                 // wave-uniform: EXEC all-ones for WMMA
    const int nBase = blockIdx.y * 64;
    const int half  = lane >> 4;            // A layout: lanes0-15 K=0,1 / lanes16-31 K=2,3
    const int li    = lane & 15;

    const float* arow = A + (size_t)(mBase + li) * (size_t)K;
    v8f acc[4] = {};
    for (int k0 = 0; k0 < K; k0 += 4) {
        const int kk = k0 + 2 * half;
        v2f a = *(const v2f*)(arow + kk);
        const float* bp = B + (size_t)kk * HID + nBase + li;
#pragma unroll
        for (int t = 0; t < 4; ++t) {
            v2f b;
            b.x = bp[t * 16];
            b.y = bp[t * 16 + HID];
            acc[t] = __builtin_amdgcn_wmma_f32_16x16x4_f32(
                false, a, false, b, (short)0, acc[t], false, false);
        }
    }
#pragma unroll
    for (int t = 0; t < 4; ++t) {
        const float bv = bias ? bias[nBase + t * 16 + li] : 0.f;
#pragma unroll
        for (int r = 0; r < 8; ++r) {
            const int rowi = mBase + r + 8 * half;          // D: lanes16-31 hold M=8..15
            const size_t idx = (size_t)rowi * HID + nBase + t * 16 + li;
            const float v = acc[t][r] + bv;
            C[idx] = v;
            if (C2) C2[idx] = v;
        }
    }
}

// ---------------- fused CSR aggregate + ReLU update (no atomics) ----------------
// one wave per node; lane owns 4 channels (float4); edge (src,w) broadcast via shfl

__global__ void graff_agg_update(const int* __restrict__ starts,
                                 const int* __restrict__ cnt,
                                 const int* __restrict__ srcIdx,
                                 const float* __restrict__ wcsr,
                                 const float* __restrict__ m,
                                 float* __restrict__ h,
                                 const float* __restrict__ x0,
                                 const float* __restrict__ omega,
                                 const float* __restrict__ betaPtr, int n) {
    const int node = (blockIdx.x * blockDim.x + threadIdx.x) >> 5;
    const int lane = threadIdx.x & 31;
    if (node >= n) return;
    const int start = starts[node];
    const int end   = start + cnt[node];
    const float4* m4 = (const float4*)m;

    float4 acc = make_float4(0.f, 0.f, 0.f, 0.f);
    for (int base = start; base < end; base += 32) {
        int c32 = end - base; if (c32 > 32) c32 = 32;
        int s = 0; float wv = 0.f;
        if (lane < c32) { s = srcIdx[base + lane]; wv = wcsr[base + lane]; }
        for (int j = 0; j < c32; ++j) {
            int   ss = __shfl(s, j);
            float ww = __shfl(wv, j);
            float4 mv = m4[(size_t)ss * 32 + lane];   // coalesced 512B row
            acc.x += ww * mv.x; acc.y += ww * mv.y;
            acc.z += ww * mv.z; acc.w += ww * mv.w;
        }
    }

    const float beta = *betaPtr;
    const size_t vi = (size_t)node * 32 + lane;
    float4 hv = ((const float4*)h)[vi];
    float4 xv = ((const float4*)x0)[vi];
    float4 ov = ((const float4*)omega)[lane];
    float4 r;
    r.x = hv.x + fmaxf(acc.x - hv.x * ov.x - beta * xv.x, 0.f);
    r.y = hv.y + fmaxf(acc.y - hv.y * ov.y - beta * xv.y, 0.f);
    r.z = hv.z + fmaxf(acc.z - hv.z * ov.z - beta * xv.z, 0.f);
    r.w = hv.w + fmaxf(acc.w - hv.w * ov.w - beta * xv.w, 0.f);
    ((float4*)h)[vi] = r;
}

// ---------------- decoder: out[n] = dot(h[n,:], W_dec) + b_dec ----------------

__global__ void graff_decoder(const float* __restrict__ h,
                              const float* __restrict__ Wd,
                              const float* __restrict__ bd,
                              float* __restrict__ out, int n) {
    const int wave = (blockIdx.x * blockDim.x + threadIdx.x) >> 5;
    const int lane = threadIdx.x & 31;
    if (wave >= n) return;
    const float* hr = h + (size_t)wave * HID;
    float s = 0.f;
#pragma unroll
    for (int j = 0; j < 4; ++j) s += hr[lane + 32 * j] * Wd[lane + 32 * j];
#pragma unroll
    for (int off = 16; off >= 1; off >>= 1) s += __shfl_xor(s, off, 32);
    if (lane == 0) out[wave] = s + bd[0];
}

// ---------------- host orchestration ----------------

extern "C" void kernel_launch(void* const* d_in, const int* in_sizes, int n_in,
                              void* d_out, int out_size, void* d_ws, size_t ws_size,
                              hipStream_t stream) {
    (void)in_sizes; (void)n_in; (void)out_size; (void)ws_size;

    const float*      x    = (const float*)d_in[0];
    const long long*  eidx = (const long long*)d_in[1];   // int64 [2, E]
    const long long*  row  = eidx;
    const long long*  col  = eidx + N_EDGES;
    const float*      ew   = (const float*)d_in[2];
    const float*      Wenc = (const float*)d_in[3];
    const float*      benc = (const float*)d_in[4];
    const float*      W    = (const float*)d_in[5];
    const float*      omg  = (const float*)d_in[6];
    const float*      beta = (const float*)d_in[7];       // device scalar
    const float*      Wdec = (const float*)d_in[8];
    const float*      bdec = (const float*)d_in[9];
    float*            out  = (float*)d_out;

    const size_t NH = (size_t)N_NODES * HID;
    char* p = (char*)d_ws;
    float* deg      = (float*)p; p += (size_t)N_NODES * 4;
    float* dinv     = (float*)p; p += (size_t)N_NODES * 4;
    int*   cnt      = (int*)p;   p += (size_t)N_NODES * 4;
    int*   starts   = (int*)p;   p += (size_t)N_NODES * 4;
    int*   cursor   = (int*)p;   p += (size_t)N_NODES * 4;
    int*   bsums    = (int*)p;   p += 1024 * 4;
    int*   srcIdx   = (int*)p;   p += (size_t)N_EDGES * 4;
    float* wcsr     = (float*)p; p += (size_t)N_EDGES * 4;
    float* Wsym     = (float*)p; p += (size_t)HID * HID * 4;
    float* x0       = (float*)p; p += NH * 4;
    float* h        = (float*)p; p += NH * 4;
    float* m        = (float*)p; p += NH * 4;

    const int scanBlocks = (N_NODES + 255) / 256;   // 391

    // ---- one-time preprocessing (per launch) ----
    hipMemsetAsync(deg,    0, (size_t)N_NODES * 4, stream);
    hipMemsetAsync(cnt,    0, (size_t)N_NODES * 4, stream);
    hipMemsetAsync(cursor, 0, (size_t)N_NODES * 4, stream);

    graff_deg_count<<<(N_EDGES + 255) / 256, 256, 0, stream>>>(col, ew, deg, cnt, N_EDGES);
    graff_dinv<<<scanBlocks, 256, 0, stream>>>(deg, dinv, N_NODES);
    graff_scan_block<<<scanBlocks, 256, 0, stream>>>(cnt, starts, bsums, N_NODES);
    graff_scan_tops<<<1, 32, 0, stream>>>(bsums, scanBlocks);
    graff_scan_add<<<scanBlocks, 256, 0, stream>>>(starts, bsums, N_NODES);
    graff_fill<<<(N_EDGES + 255) / 256, 256, 0, stream>>>(row, col, ew, dinv, starts,
                                                          cursor, srcIdx, wcsr, N_EDGES);
    graff_wsym<<<(HID * HID) / 256, 256, 0, stream>>>(W, Wsym);

    // ---- encoder: x0 = x @ W_enc + b_enc ; h = x0 ----
    const int mTiles = N_NODES / 16;                // 6250 (exact)
    dim3 ggrid((mTiles + 7) / 8, HID / 64);         // (782, 2)
    graff_wmma_gemm<<<ggrid, 256, 0, stream>>>(x, Wenc, benc, x0, h, N_NODES, IN_CH);

    // ---- 8 shared-weight layers: GEMM (WMMA) + fused gather/update ----
    for (int l = 0; l < NUM_LAYERS; ++l) {
        graff_wmma_gemm<<<ggrid, 256, 0, stream>>>(h, Wsym, nullptr, m, nullptr, N_NODES, HID);
        graff_agg_update<<<N_NODES / 8, 256, 0, stream>>>(starts, cnt, srcIdx, wcsr,
                                                          m, h, x0, omg, beta, N_NODES);
    }

    // ---- decoder ----
    graff_decoder<<<(N_NODES + 7) / 8, 256, 0, stream>>>(h, Wdec, bdec, out, N_NODES);
}